// Attention_50826642981654
// MI455X (gfx1250) — compile-verified
//
#include <hip/hip_runtime.h>

#define SEQL   2048
#define DIMM   2048
#define NHEADS 16
#define HDIM   128

typedef float v2f __attribute__((ext_vector_type(2)));
typedef float v8f __attribute__((ext_vector_type(8)));

// D = A(16x4) * B(4x16) + C, fp32 WMMA
__device__ __forceinline__ v8f wmma_f32(v2f a, v2f b, v8f c) {
    return __builtin_amdgcn_wmma_f32_16x16x4_f32(false, a, false, b, (short)0, c, false, false);
}

// ---------------------------------------------------------------------------
// C[M,N] = A[M,K] * B[N,K]^T   (row-major A, row-major B; i.e. x @ W.T)
// M=N=K=2048. Block = 256 threads = 8 waves. Block tile 128(M) x 64(N).
// Each wave: 16(M) x 64(N), A-fragment reused across 4 N-subtiles.
// ---------------------------------------------------------------------------
__global__ __launch_bounds__(256)
void gemm_abt(const float* __restrict__ A, const float* __restrict__ B,
              float* __restrict__ C) {
    const int lane = threadIdx.x & 31;
    const int wave = threadIdx.x >> 5;
    const int hi   = lane >> 4;     // half-wave select
    const int lc   = lane & 15;

    const int m0 = blockIdx.y * 128 + wave * 16;
    const int n0 = blockIdx.x * 64;

    v8f acc0 = {}, acc1 = {}, acc2 = {}, acc3 = {};

    const float* ap  = A + (size_t)(m0 + lc) * DIMM + 2 * hi;
    const float* bp0 = B + (size_t)(n0 +  0 + lc) * DIMM + 2 * hi;
    const float* bp1 = B + (size_t)(n0 + 16 + lc) * DIMM + 2 * hi;
    const float* bp2 = B + (size_t)(n0 + 32 + lc) * DIMM + 2 * hi;
    const float* bp3 = B + (size_t)(n0 + 48 + lc) * DIMM + 2 * hi;

#pragma unroll 4
    for (int k0 = 0; k0 < DIMM; k0 += 4) {
        v2f a  = *(const v2f*)(ap  + k0);
        v2f b0 = *(const v2f*)(bp0 + k0);
        v2f b1 = *(const v2f*)(bp1 + k0);
        v2f b2 = *(const v2f*)(bp2 + k0);
        v2f b3 = *(const v2f*)(bp3 + k0);
        acc0 = wmma_f32(a, b0, acc0);
        acc1 = wmma_f32(a, b1, acc1);
        acc2 = wmma_f32(a, b2, acc2);
        acc3 = wmma_f32(a, b3, acc3);
    }

#pragma unroll
    for (int r = 0; r < 8; ++r) {
        float* crow = C + (size_t)(m0 + r + 8 * hi) * DIMM + n0 + lc;
        crow[ 0] = acc0[r];
        crow[16] = acc1[r];
        crow[32] = acc2[r];
        crow[48] = acc3[r];
    }
}

// ---------------------------------------------------------------------------
// Flash-style attention per head, fp32 WMMA.
// grid = (SEQ/128, NHEADS), block = 256 (8 waves). Wave = one 16-query tile.
// Q/K/V are [SEQ, DIM] with head h in columns [h*128, h*128+128).
// ---------------------------------------------------------------------------
__global__ __launch_bounds__(256)
void attn_fwd(const float* __restrict__ Q, const float* __restrict__ K,
              const float* __restrict__ V, const float* __restrict__ mask,
              float* __restrict__ O) {
    __shared__ float ldsP[8 * 256];   // per-wave 16x16 P scratch

    const int lane = threadIdx.x & 31;
    const int wave = threadIdx.x >> 5;
    const int hi   = lane >> 4;
    const int lc   = lane & 15;

    const int h  = blockIdx.y;
    const int q0 = (blockIdx.x * 8 + wave) * 16;

    // Q tile -> 32 A-fragments (d = 0..127 in steps of 4)
    v2f aq[32];
    {
        const float* qrow = Q + (size_t)(q0 + lc) * DIMM + h * HDIM + 2 * hi;
#pragma unroll
        for (int s = 0; s < 32; ++s) aq[s] = *(const v2f*)(qrow + s * 4);
    }

    v8f oacc[8];
#pragma unroll
    for (int t = 0; t < 8; ++t) oacc[t] = (v8f){};
    float mrow[8], lrow[8];
#pragma unroll
    for (int r = 0; r < 8; ++r) { mrow[r] = -1.0e30f; lrow[r] = 0.0f; }

    float* lp = &ldsP[wave * 256];
    const float scale = 0.08838834764831845f;   // 1/sqrt(128)
    const size_t mbase = ((size_t)h * SEQL + q0) * SEQL;

#pragma unroll 1
    for (int kv0 = 0; kv0 < SEQL; kv0 += 16) {
        // ---- S = Q * K^T  (16x16 tile) ----
        v8f s = {};
        const float* krow = K + (size_t)(kv0 + lc) * DIMM + h * HDIM + 2 * hi;
#pragma unroll
        for (int ss = 0; ss < 32; ++ss) {
            v2f bk = *(const v2f*)(krow + ss * 4);
            s = wmma_f32(aq[ss], bk, s);
        }

        // ---- scale + mask (non-temporal: don't pollute L2), online softmax ----
#pragma unroll
        for (int r = 0; r < 8; ++r) {
            const float mval = __builtin_nontemporal_load(
                mask + mbase + (size_t)(r + 8 * hi) * SEQL + kv0 + lc);
            float v = s[r] * scale + mval;

            float rmax = v;
            rmax = fmaxf(rmax, __shfl_xor(rmax, 1, 32));
            rmax = fmaxf(rmax, __shfl_xor(rmax, 2, 32));
            rmax = fmaxf(rmax, __shfl_xor(rmax, 4, 32));
            rmax = fmaxf(rmax, __shfl_xor(rmax, 8, 32));

            const float mnew = fmaxf(mrow[r], rmax);
            const float corr = __expf(mrow[r] - mnew);
            mrow[r] = mnew;

            const float p = __expf(v - mnew);
            s[r] = p;

            float psum = p;
            psum += __shfl_xor(psum, 1, 32);
            psum += __shfl_xor(psum, 2, 32);
            psum += __shfl_xor(psum, 4, 32);
            psum += __shfl_xor(psum, 8, 32);
            lrow[r] = lrow[r] * corr + psum;

#pragma unroll
            for (int t = 0; t < 8; ++t) oacc[t][r] *= corr;
        }

        // ---- P: D-layout -> A-layout via per-wave LDS scratch ----
#pragma unroll
        for (int r = 0; r < 8; ++r) lp[(r + 8 * hi) * 16 + lc] = s[r];
        asm volatile("s_wait_dscnt 0" ::: "memory");    // RAW across layout swap
        v2f pa[4];
#pragma unroll
        for (int kk = 0; kk < 4; ++kk)
            pa[kk] = *(const v2f*)&lp[lc * 16 + kk * 4 + 2 * hi];
        asm volatile("s_wait_dscnt 0" ::: "memory");    // loads done before reuse

        // ---- O += P * V  (16x16 P times 16x128 V) ----
#pragma unroll
        for (int kk = 0; kk < 4; ++kk) {
            const float* vrow =
                V + (size_t)(kv0 + kk * 4 + 2 * hi) * DIMM + h * HDIM + lc;
#pragma unroll
            for (int t = 0; t < 8; ++t) {
                v2f bv;
                bv[0] = vrow[t * 16];
                bv[1] = vrow[t * 16 + DIMM];
                oacc[t] = wmma_f32(pa[kk], bv, oacc[t]);
            }
        }
    }

    // ---- normalize and store ----
#pragma unroll
    for (int r = 0; r < 8; ++r) {
        const float inv = 1.0f / lrow[r];
        float* orow = O + (size_t)(q0 + r + 8 * hi) * DIMM + h * HDIM + lc;
#pragma unroll
        for (int t = 0; t < 8; ++t) orow[t * 16] = oacc[t][r] * inv;
    }
}

// ---------------------------------------------------------------------------
extern "C" void kernel_launch(void* const* d_in, const int* in_sizes, int n_in,
                              void* d_out, int out_size, void* d_ws, size_t ws_size,
                              hipStream_t stream) {
    const float* x    = (const float*)d_in[0];
    const float* mask = (const float*)d_in[1];
    // d_in[2] = start_pos (always 0 for this prefill shape)
    const float* wq   = (const float*)d_in[3];
    const float* wk   = (const float*)d_in[4];
    const float* wv   = (const float*)d_in[5];
    const float* wo   = (const float*)d_in[6];
    float* out = (float*)d_out;

    float* ws = (float*)d_ws;
    const size_t MAT = (size_t)SEQL * DIMM;   // 4M floats = 16 MB
    float* Qb = ws;
    float* Kb = ws + MAT;
    float* Vb = ws + 2 * MAT;
    float* Ab = ws + 3 * MAT;

    dim3 tb(256);
    dim3 gg(DIMM / 64, SEQL / 128);           // 32 x 16 blocks
    gemm_abt<<<gg, tb, 0, stream>>>(x, wq, Qb);
    gemm_abt<<<gg, tb, 0, stream>>>(x, wk, Kb);
    gemm_abt<<<gg, tb, 0, stream>>>(x, wv, Vb);

    dim3 ga(SEQL / 128, NHEADS);              // 16 x 16 blocks
    attn_fwd<<<ga, tb, 0, stream>>>(Qb, Kb, Vb, mask, Ab);

    gemm_abt<<<gg, tb, 0, stream>>>(Ab, wo, out);
}